// ProbeAggregation_84164179132609
// MI455X (gfx1250) — compile-verified
//
#include <hip/hip_runtime.h>
#include <hip/hip_bf16.h>

typedef __attribute__((ext_vector_type(2))) float v2f;
typedef __attribute__((ext_vector_type(8))) float v8f;

#define B_N   4
#define T_N   4096
#define P_N   16384
#define H_N   256
#define K_NN  5
#define TILE  16
#define WAVES 4
#define BLOCK (WAVES * 32)

// Branchless insert of (x, xi) into ascending sorted 5-element list.
__device__ __forceinline__ void insert5(float x, int xi, float d[K_NN], int id[K_NN]) {
    const bool c0 = x < d[0], c1 = x < d[1], c2 = x < d[2], c3 = x < d[3], c4 = x < d[4];
    d[4]  = c4 ? (c3 ? d[3]  : x ) : d[4];
    id[4] = c4 ? (c3 ? id[3] : xi) : id[4];
    d[3]  = c3 ? (c2 ? d[2]  : x ) : d[3];
    id[3] = c3 ? (c2 ? id[2] : xi) : id[3];
    d[2]  = c2 ? (c1 ? d[1]  : x ) : d[2];
    id[2] = c2 ? (c1 ? id[1] : xi) : id[2];
    d[1]  = c1 ? (c0 ? d[0]  : x ) : d[1];
    id[1] = c1 ? (c0 ? id[0] : xi) : id[1];
    d[0]  = c0 ? x  : d[0];
    id[0] = c0 ? xi : id[0];
}

__global__ __launch_bounds__(BLOCK)
void ProbeAggregation_84164179132609_kernel(const float* __restrict__ emb,
                                            const float* __restrict__ tpos,
                                            const float* __restrict__ ppos,
                                            float* __restrict__ out) {
    __shared__ float sD[WAVES][32][8][K_NN];
    __shared__ int   sI[WAVES][32][8][K_NN];
    __shared__ int   sTop[WAVES][TILE][K_NN];

    const int lane = threadIdx.x & 31;
    const int wv   = threadIdx.x >> 5;
    const int gw   = blockIdx.x * WAVES + wv;          // global wave id = probe tile id
    const int tilesPerB = P_N / TILE;                  // 1024
    const int b     = gw / tilesPerB;
    const int pbase = (gw % tilesPerB) * TILE;
    const int m     = lane & 15;
    const bool lo   = lane < 16;

    // ---- Build A (16x4 f32, 2 VGPRs): row = (px, py, |p|^2, 1) ----
    // ISA layout: lanes 0-15 hold K=0 (v0), K=1 (v1); lanes 16-31 hold K=2 (v0), K=3 (v1).
    const float2 pp = ((const float2*)ppos)[(size_t)b * P_N + pbase + m];
    const float  sp = pp.x * pp.x + pp.y * pp.y;
    v2f A;
    A.x = lo ? pp.x : sp;
    A.y = lo ? pp.y : 1.0f;

    // Per-lane running top-5 per accumulator slot (8 slots -> 8 probe rows).
    float bd[8][K_NN];
    int   bi[8][K_NN];
    #pragma unroll
    for (int r = 0; r < 8; ++r)
        #pragma unroll
        for (int j = 0; j < K_NN; ++j) { bd[r][j] = 3.0e38f; bi[r][j] = 0; }

    const float2* tp = ((const float2*)tpos) + (size_t)b * T_N;

    // ---- Distance sweep: 256 WMMAs produce all 16 x 4096 squared distances ----
    for (int tbase = 0; tbase < T_N; tbase += TILE) {
        const float2 tq = tp[tbase + m];
        const float  st = tq.x * tq.x + tq.y * tq.y;
        v2f Bm;                                        // column = (-2tx, -2ty, 1, |t|^2)
        Bm.x = lo ? (-2.0f * tq.x) : 1.0f;
        Bm.y = lo ? (-2.0f * tq.y) : st;
        v8f C = {};
        // d^2[M][N] = A row(M) . B col(N), exact in f32
        C = __builtin_amdgcn_wmma_f32_16x16x4_f32(false, A, false, Bm,
                                                  (short)0, C, false, false);
        const int tidx = tbase + m;                    // turbine index this lane sees
        #pragma unroll
        for (int r = 0; r < 8; ++r)
            insert5(C[r], tidx, bd[r], bi[r]);
    }

    // ---- Dump per-lane candidates to LDS ----
    #pragma unroll
    for (int r = 0; r < 8; ++r)
        #pragma unroll
        for (int j = 0; j < K_NN; ++j) {
            sD[wv][lane][r][j] = bd[r][j];
            sI[wv][lane][r][j] = bi[r][j];
        }
    __syncthreads();

    // ---- Final top-5 per probe: lane m reduces probe m's 16x5 candidates ----
    // C layout: slot r, lanes 0-15 -> probe M=r; lanes 16-31 -> probe M=r+8.
    if (lane < TILE) {
        float fd[K_NN]; int fi[K_NN];
        #pragma unroll
        for (int j = 0; j < K_NN; ++j) { fd[j] = 3.0e38f; fi[j] = 0; }
        const int base = (lane < 8) ? 0 : 16;
        const int slot = lane & 7;
        for (int l = 0; l < 16; ++l)
            #pragma unroll
            for (int j = 0; j < K_NN; ++j)
                insert5(sD[wv][base + l][slot][j], sI[wv][base + l][slot][j], fd, fi);
        #pragma unroll
        for (int j = 0; j < K_NN; ++j) sTop[wv][lane][j] = fi[j];
    }
    __syncthreads();

    // ---- Gather 5 embeddings per probe (L2-resident), mean, coalesced store ----
    const float4* eb4 = (const float4*)(emb + (size_t)b * T_N * H_N);
    float4*       ob4 = (float4*)(out + ((size_t)b * P_N + pbase) * (size_t)H_N);
    const int H4 = H_N / 4;                            // 64 float4 per row

    for (int pm = 0; pm < TILE; ++pm) {
        const int i0 = sTop[wv][pm][0];
        const int i1 = sTop[wv][pm][1];
        const int i2 = sTop[wv][pm][2];
        const int i3 = sTop[wv][pm][3];
        const int i4 = sTop[wv][pm][4];
        #pragma unroll
        for (int c = 0; c < 2; ++c) {
            const int h4 = lane + c * 32;
            const float4 a0 = eb4[(size_t)i0 * H4 + h4];
            const float4 a1 = eb4[(size_t)i1 * H4 + h4];
            const float4 a2 = eb4[(size_t)i2 * H4 + h4];
            const float4 a3 = eb4[(size_t)i3 * H4 + h4];
            const float4 a4 = eb4[(size_t)i4 * H4 + h4];
            float4 s;
            s.x = (a0.x + a1.x + a2.x + a3.x + a4.x) * 0.2f;
            s.y = (a0.y + a1.y + a2.y + a3.y + a4.y) * 0.2f;
            s.z = (a0.z + a1.z + a2.z + a3.z + a4.z) * 0.2f;
            s.w = (a0.w + a1.w + a2.w + a3.w + a4.w) * 0.2f;
            ob4[(size_t)pm * H4 + h4] = s;
        }
    }
}

extern "C" void kernel_launch(void* const* d_in, const int* in_sizes, int n_in,
                              void* d_out, int out_size, void* d_ws, size_t ws_size,
                              hipStream_t stream) {
    const float* emb  = (const float*)d_in[0];   // [B, T, H]
    const float* tpos = (const float*)d_in[1];   // [B, T, 2]
    const float* ppos = (const float*)d_in[2];   // [B, P, 2]
    float* out = (float*)d_out;                  // [B, P, H]

    const int totalWaves = B_N * (P_N / TILE);   // 4096 probe tiles
    dim3 grid(totalWaves / WAVES);               // 1024 blocks
    dim3 block(BLOCK);                           // 128 threads = 4 waves
    hipLaunchKernelGGL(ProbeAggregation_84164179132609_kernel, grid, block, 0, stream,
                       emb, tpos, ppos, out);
}